// ModelNew_36215164240272
// MI455X (gfx1250) — compile-verified
//
#include <hip/hip_runtime.h>

// ---------------------------------------------------------------------------
// Mamba-2 SSD chunked scan for MI455X (gfx1250, wave32, WMMA f32 16x16x4).
// b=512, s=128, h=8, p=64, n=16, BLOCK_LEN=64 -> 2 chunks per (b,h).
// One workgroup (256 threads = 8 wave32) per (b,h); chunks processed in
// sequence so chunk-0 "states" (LDS) feed chunk-1 Y_off.
// Memory-bound problem (~340 MB @ 23.3 TB/s => ~15 us floor), so full-fp32
// WMMA is the right precision; GEMMs go to the matrix pipe, exp/segsum
// elementwise work stays on VALU (factored so only 3x64 exps per chunk).
// ---------------------------------------------------------------------------

typedef float v2f __attribute__((ext_vector_type(2)));
typedef float v8f __attribute__((ext_vector_type(8)));

#define XSTR 68   // LDS row stride for 64x64 tiles (16B aligned, conflict-free)
#define NSTR 20   // LDS row stride for 64x16 tiles (16B aligned, conflict-free)

static __device__ __forceinline__ v8f wmma_f32(v2f a, v2f b, v8f c) {
  // D = A(16x4 f32) * B(4x16 f32) + C(16x16 f32)
  return __builtin_amdgcn_wmma_f32_16x16x4_f32(false, a, false, b, (short)0, c,
                                               false, false);
}

__global__ __launch_bounds__(256) void ssd_mamba2_kernel(
    const float* __restrict__ Xg,  // (b,128,8,64)
    const float* __restrict__ Ag,  // (b,128,8)
    const float* __restrict__ Bg,  // (b,128,8,16)
    const float* __restrict__ Cg,  // (b,128,8,16)
    float* __restrict__ Yg)        // (b,128,8,64)
{
  __shared__ float Xs[64 * XSTR];  // X chunk   (l x p)
  __shared__ float Ms[64 * XSTR];  // (C B^T) o L  (l x l)
  __shared__ float Bs[64 * NSTR];  // B chunk   (l x n)
  __shared__ float Cs[64 * NSTR];  // C chunk   (l x n)
  __shared__ float Ss[64 * NSTR];  // states    (p x n), persists chunk0->1
  __shared__ float AcsS[64];       // inclusive cumsum of A chunk
  __shared__ float AexS[64];       // exclusive cumsum of A chunk
  __shared__ float dS[64];         // A values, then exp(S_c - Acs)
  __shared__ float eA[64];         // exp(Acs[i])
  __shared__ float eX[64];         // exp(-Aexcl[s])
  __shared__ float sMeta[2];       // [0] = S0, [1] = exp(S0)

  const int tid  = threadIdx.x;
  const int b    = blockIdx.x >> 3;  // / h
  const int h    = blockIdx.x & 7;
  const int lane = tid & 31;
  const int lm   = lane & 15;        // N / M-low index inside fragment
  const int hh   = lane >> 4;        // half-wave select
  const int wv   = __builtin_amdgcn_readfirstlane(tid >> 5);  // scalar wave id

  for (int cc = 0; cc < 2; ++cc) {
    // ---- A chunk: load + serial cumsum (64 elements, trivial) ------------
    if (tid < 64) dS[tid] = Ag[(b * 128 + cc * 64 + tid) * 8 + h];
    if (cc == 0 && tid < 64) {
      // prefetch chunk-1 X rows into L2 while we work on chunk 0
      __builtin_prefetch(Xg + ((size_t)(b * 128 + 64 + tid) * 8 + h) * 64, 0, 1);
    }
    __syncthreads();
    if (tid < 64) {
      float s = 0.f;
      for (int k = 0; k <= tid; ++k) s += dS[k];
      AcsS[tid] = s;
      AexS[tid] = s - dS[tid];
    }
    __syncthreads();
    const float Slast = AcsS[63];
    if (tid < 64) {
      const float acs = AcsS[tid];
      dS[tid] = __expf(Slast - acs);   // decay_states
      eA[tid] = __expf(acs);           // segsum row factor / Y_off scale
      eX[tid] = __expf(-AexS[tid]);    // segsum col factor
    }
    if (cc == 0 && tid == 0) {
      sMeta[0] = Slast;
      sMeta[1] = __expf(Slast);
    }

    // ---- stage X, B, C in LDS (float4, coalesced) ------------------------
    {
      const int xbase = ((b * 128 + cc * 64) * 8 + h) * 64;
      for (int e = tid; e < 64 * 16; e += 256) {  // 64 rows x 16 float4
        const int r = e >> 4, q = (e & 15) << 2;
        const float4 v =
            *reinterpret_cast<const float4*>(Xg + xbase + r * 512 + q);
        *reinterpret_cast<float4*>(&Xs[r * XSTR + q]) = v;
      }
      const int nbase = ((b * 128 + cc * 64) * 8 + h) * 16;
      const int r = tid >> 2, q = (tid & 3) << 2;  // 64 rows x 4 float4
      *reinterpret_cast<float4*>(&Bs[r * NSTR + q]) =
          *reinterpret_cast<const float4*>(Bg + nbase + r * 128 + q);
      *reinterpret_cast<float4*>(&Cs[r * NSTR + q]) =
          *reinterpret_cast<const float4*>(Cg + nbase + r * 128 + q);
    }
    __syncthreads();

    // ---- G = C * B^T (K=16), then M = G o L -> Ms ------------------------
    // Each wave owns subtiles t = 2*wv, 2*wv+1 of the 4x4 grid.
    for (int t = 2 * wv; t <= 2 * wv + 1; ++t) {
      const int mi = t >> 2, ni = t & 3;
      if (ni > mi) {  // strictly above diagonal: M block is zero
        for (int v = 0; v < 8; ++v)
          Ms[(16 * mi + v + 8 * hh) * XSTR + 16 * ni + lm] = 0.f;
      } else {
        v8f acc = {};
#pragma unroll
        for (int kk = 0; kk < 4; ++kk) {
          const int k0 = 4 * kk + 2 * hh;
          v2f a, bf;
          a.x  = Cs[(16 * mi + lm) * NSTR + k0];
          a.y  = Cs[(16 * mi + lm) * NSTR + k0 + 1];
          bf.x = Bs[(16 * ni + lm) * NSTR + k0];   // B^T[k][s] = B[s][k]
          bf.y = Bs[(16 * ni + lm) * NSTR + k0 + 1];
          acc = wmma_f32(a, bf, acc);
        }
        // L[i,s] = exp(Acs[i]) * exp(-Aexcl[s]) for s<=i, else 0.
        // Factored exps are precomputed -> per element: 1 LDS bcast + mul +
        // cndmask (no transcendental, no exec-mask branching here).
        const int s = 16 * ni + lm;  // col (l index), constant per lane
        const float eXs = eX[s];
#pragma unroll
        for (int v = 0; v < 8; ++v) {
          const int i = 16 * mi + v + 8 * hh;  // row (l index)
          float val = acc[v] * eA[i] * eXs;
          if (s > i) val = 0.f;                // diagonal subtile mask only
          Ms[i * XSTR + s] = val;
        }
      }
    }
    __syncthreads();

    // ---- chunk-0 states: states[p,n] = sum_s X[s,p]*d[s]*B[s,n] ----------
    if (cc == 0 && wv < 4) {  // scalar branch: EXEC stays all-ones for WMMA
      const int mi = wv;      // p block
      v8f acc = {};
#pragma unroll 4
      for (int kk = 0; kk < 16; ++kk) {
        const int k0 = 4 * kk + 2 * hh;
        v2f a, bf;
        a.x  = Xs[k0 * XSTR + 16 * mi + lm] * dS[k0];        // A[p][s]
        a.y  = Xs[(k0 + 1) * XSTR + 16 * mi + lm] * dS[k0 + 1];
        bf.x = Bs[k0 * NSTR + lm];                           // B[s][n]
        bf.y = Bs[(k0 + 1) * NSTR + lm];
        acc = wmma_f32(a, bf, acc);
      }
#pragma unroll
      for (int v = 0; v < 8; ++v)
        Ss[(16 * mi + v + 8 * hh) * NSTR + lm] = acc[v];
    }

    // ---- Y = M @ X  (+ chunk-1 off-diagonal term), store -----------------
    for (int t = 2 * wv; t <= 2 * wv + 1; ++t) {
      const int mi = t >> 2, pi = t & 3;
      v8f acc = {};
      const int kkEnd = 4 * (mi + 1);  // M rows are lower-triangular
#pragma unroll 4
      for (int kk = 0; kk < kkEnd; ++kk) {
        const int k0 = 4 * kk + 2 * hh;
        v2f a, bf;
        a.x  = Ms[(16 * mi + lm) * XSTR + k0];
        a.y  = Ms[(16 * mi + lm) * XSTR + k0 + 1];
        bf.x = Xs[k0 * XSTR + 16 * pi + lm];
        bf.y = Xs[(k0 + 1) * XSTR + 16 * pi + lm];
        acc = wmma_f32(a, bf, acc);
      }
      if (cc == 1) {
        // Y_off = diag(exp(Acs1[i] + S0)) * C1 * states0^T   (K = n = 16)
        const float expS0 = sMeta[1];
        v8f off = {};
#pragma unroll
        for (int kk = 0; kk < 4; ++kk) {
          const int k0 = 4 * kk + 2 * hh;
          v2f a, bf;
          a.x  = Cs[(16 * mi + lm) * NSTR + k0];
          a.y  = Cs[(16 * mi + lm) * NSTR + k0 + 1];
          bf.x = Ss[(16 * pi + lm) * NSTR + k0];  // states^T[n][p]
          bf.y = Ss[(16 * pi + lm) * NSTR + k0 + 1];
          off = wmma_f32(a, bf, off);
        }
#pragma unroll
        for (int v = 0; v < 8; ++v) {
          const int i = 16 * mi + v + 8 * hh;
          acc[v] += off[v] * (eA[i] * expS0);  // exp(Acs1[i] + S0)
        }
      }
      const int obase = ((b * 128 + cc * 64) * 8 + h) * 64 + 16 * pi + lm;
#pragma unroll
      for (int v = 0; v < 8; ++v) {
        const int i = 16 * mi + v + 8 * hh;
        Yg[obase + i * 512] = acc[v];
      }
    }
    __syncthreads();  // protect LDS (incl. Ss) before next chunk reloads
  }
}

extern "C" void kernel_launch(void* const* d_in, const int* in_sizes, int n_in,
                              void* d_out, int out_size, void* d_ws,
                              size_t ws_size, hipStream_t stream) {
  (void)in_sizes; (void)n_in; (void)d_ws; (void)ws_size; (void)out_size;
  const float* X = (const float*)d_in[0];
  const float* A = (const float*)d_in[1];
  const float* B = (const float*)d_in[2];
  const float* C = (const float*)d_in[3];
  float* Y = (float*)d_out;
  // 512 batches * 8 heads = 4096 workgroups, 256 threads (8 wave32) each
  ssd_mamba2_kernel<<<dim3(4096), dim3(256), 0, stream>>>(X, A, B, C, Y);
}